// GumbelSinkhorn_46591805227451
// MI455X (gfx1250) — compile-verified
//
#include <hip/hip_runtime.h>
#include <hip/hip_bf16.h>
#include <math.h>

// Problem constants (from reference): B=64, N=512, 60 Sinkhorn iterations.
#define NB    64
#define NN    512
#define ITERS 60
#define TEMP_INV (1.0f / 3.0f)
#define GAMMA 0.01f
#define ROWS_PER_BLOCK 8            // 256 threads = 8 waves, one wave per row
#define BLOCKS_PER_BATCH (NN / ROWS_PER_BLOCK)   // 64

typedef float v2f __attribute__((ext_vector_type(2)));
typedef float v4f __attribute__((ext_vector_type(4)));
typedef float v8f __attribute__((ext_vector_type(8)));

// ---- wave32 tree reductions ----
__device__ __forceinline__ float waveSum(float v) {
    #pragma unroll
    for (int off = 16; off > 0; off >>= 1) v += __shfl_xor(v, off, 32);
    return v;
}
__device__ __forceinline__ float waveMax(float v) {
    #pragma unroll
    for (int off = 16; off > 0; off >>= 1) v = fmaxf(v, __shfl_xor(v, off, 32));
    return v;
}

// Block-level partial column sums: each of the 8 waves stages its row (the
// freshly computed output values) into LDS, then 256 threads reduce the 8 rows
// in a FIXED order (deterministic fp) and write one 512-float partial per block.
__device__ __forceinline__ void emitPartial(float (*sh)[NN], const v4f* o,
                                            int wave, int lane,
                                            float* __restrict__ partialRow) {
    v4f* shr = (v4f*)sh[wave];
    #pragma unroll
    for (int k = 0; k < 4; ++k) shr[lane + 32 * k] = o[k];
    __syncthreads();
    #pragma unroll
    for (int c = threadIdx.x; c < NN; c += 256) {
        float s = 0.0f;
        #pragma unroll
        for (int w = 0; w < ROWS_PER_BLOCK; ++w) s += sh[w][c];  // fixed order
        partialRow[c] = s;
    }
}

// ---------------------------------------------------------------------------
// Kernel 1: y = (logits + GAMMA * (-log(-log(u)))) / T ; P = softmax_row(y).
// Inputs are single-use: stream them with non-temporal loads so the 128 MB of
// logits+u does not evict the L2-resident P. Also emits block partial colsums
// of P0 so iteration 1 needs no separate full-matrix reduction pass.
// ---------------------------------------------------------------------------
__global__ void gs_init_softmax_emit(const float* __restrict__ logits,
                                     const float* __restrict__ u,
                                     float* __restrict__ P,
                                     float* __restrict__ partials) {
    __shared__ float sh[ROWS_PER_BLOCK][NN];     // 16 KB
    const int wave = threadIdx.x >> 5;
    const int lane = threadIdx.x & 31;
    const int row  = blockIdx.x * ROWS_PER_BLOCK + wave;
    const size_t row4 = (size_t)row * (NN / 4);

    const v4f* lg4 = ((const v4f*)logits) + row4;
    const v4f* uu4 = ((const v4f*)u) + row4;
    v4f* p4 = ((v4f*)P) + row4;

    float y[16];
    float m = -3.402823466e38f;
    #pragma unroll
    for (int k = 0; k < 4; ++k) {
        const v4f L = __builtin_nontemporal_load(&lg4[lane + 32 * k]);
        const v4f U = __builtin_nontemporal_load(&uu4[lane + 32 * k]);
        #pragma unroll
        for (int c = 0; c < 4; ++c) {
            const float eps = -__logf(-__logf(U[c]));
            const float yv  = (L[c] + GAMMA * eps) * TEMP_INV;
            y[4 * k + c] = yv;
            m = fmaxf(m, yv);
        }
    }
    m = waveMax(m);

    float s = 0.0f;
    #pragma unroll
    for (int i = 0; i < 16; ++i) {
        const float e = __expf(y[i] - m);
        y[i] = e;
        s += e;
    }
    s = waveSum(s);
    const float inv = 1.0f / s;

    v4f o[4];
    #pragma unroll
    for (int k = 0; k < 4; ++k) {
        #pragma unroll
        for (int c = 0; c < 4; ++c) o[k][c] = y[4 * k + c] * inv;
        p4[lane + 32 * k] = o[k];
    }
    emitPartial(sh, o, wave, lane, partials + (size_t)blockIdx.x * NN);
}

// ---------------------------------------------------------------------------
// Kernel 2 (preferred path): reduce 64 block partials per batch into colsum
// via V_WMMA_F32_16X16X4_F32 (A = ones(16x4), B = partials chunk(4x16)).
// The f32 C/D layout replicates column sums across both lane halves, so the
// result is extracted with no cross-lane shuffles. One wave per 16-col tile.
// ---------------------------------------------------------------------------
__global__ void gs_colsum_finalize_wmma(const float* __restrict__ partials,
                                        float* __restrict__ colsum) {
    const int w    = (blockIdx.x * blockDim.x + threadIdx.x) >> 5;
    const int lane = threadIdx.x & 31;
    const int b    = w >> 5;         // 32 column-tiles per batch
    const int c0   = (w & 31) << 4;

    const float* Pb = partials + (size_t)b * BLOCKS_PER_BATCH * NN;
    const int col   = c0 + (lane & 15);
    const int rsel  = (lane >> 4) << 1;   // lanes 0-15: rows p,p+1 ; 16-31: p+2,p+3

    const v2f a = {1.0f, 1.0f};
    v8f acc = {};
    #pragma unroll
    for (int p = 0; p < BLOCKS_PER_BATCH; p += 4) {
        const float* base = Pb + (size_t)(p + rsel) * NN + col;
        v2f bm;
        bm[0] = base[0];
        bm[1] = base[NN];
        acc = __builtin_amdgcn_wmma_f32_16x16x4_f32(
            false, a, false, bm, (short)0, acc, false, false);
    }
    if (lane < 16) colsum[(b << 9) + c0 + lane] = acc[0];
}

// ---------------------------------------------------------------------------
// Kernel 3 (preferred path): fused column + row normalization that ALSO emits
// block partial colsums of the matrix it just wrote, so the next iteration's
// column reduction never re-reads the full 64 MB matrix (reads 8 MB instead).
// ---------------------------------------------------------------------------
__global__ void gs_normalize_emit(float* __restrict__ P,
                                  const float* __restrict__ colsum,
                                  float* __restrict__ partials) {
    __shared__ float sh[ROWS_PER_BLOCK][NN];     // 16 KB
    const int wave = threadIdx.x >> 5;
    const int lane = threadIdx.x & 31;
    const int row  = blockIdx.x * ROWS_PER_BLOCK + wave;
    const int b    = row >> 9;

    v4f* p4 = ((v4f*)P) + (size_t)row * (NN / 4);
    const v4f* cs4 = ((const v4f*)colsum) + (size_t)b * (NN / 4);

    float q[16];
    float s = 0.0f;
    #pragma unroll
    for (int k = 0; k < 4; ++k) {
        const v4f p = p4[lane + 32 * k];
        const v4f c = cs4[lane + 32 * k];
        #pragma unroll
        for (int cc = 0; cc < 4; ++cc) {
            const float qv = p[cc] / c[cc];
            q[4 * k + cc] = qv;
            s += qv;
        }
    }
    s = waveSum(s);
    const float inv = 1.0f / s;

    v4f o[4];
    #pragma unroll
    for (int k = 0; k < 4; ++k) {
        #pragma unroll
        for (int cc = 0; cc < 4; ++cc) o[k][cc] = q[4 * k + cc] * inv;
        p4[lane + 32 * k] = o[k];
    }
    emitPartial(sh, o, wave, lane, partials + (size_t)blockIdx.x * NN);
}

// ---------------------------------------------------------------------------
// Fallback kernels (used only if d_ws is too small for the 8 MB partials):
// round-1 structure — full-matrix WMMA colsum + plain normalize.
// ---------------------------------------------------------------------------
__global__ void gs_colsum_wmma(const float* __restrict__ P,
                               float* __restrict__ colsum) {
    const int w    = (blockIdx.x * blockDim.x + threadIdx.x) >> 5;
    const int lane = threadIdx.x & 31;
    const int b    = w >> 5;
    const int c0   = (w & 31) << 4;

    const float* Pb = P + (size_t)b * NN * NN;
    const int col   = c0 + (lane & 15);
    const int rsel  = (lane >> 4) << 1;

    const v2f a = {1.0f, 1.0f};
    v8f acc = {};
    for (int r = 0; r < NN; r += 4) {
        const float* base = Pb + (size_t)(r + rsel) * NN + col;
        v2f bm;
        bm[0] = base[0];
        bm[1] = base[NN];
        acc = __builtin_amdgcn_wmma_f32_16x16x4_f32(
            false, a, false, bm, (short)0, acc, false, false);
    }
    if (lane < 16) colsum[(b << 9) + c0 + lane] = acc[0];
}

__global__ void gs_normalize(float* __restrict__ P,
                             const float* __restrict__ colsum) {
    const int wave = (blockIdx.x * blockDim.x + threadIdx.x) >> 5;
    const int lane = threadIdx.x & 31;
    const int b = wave >> 9;

    v4f* p4 = ((v4f*)P) + (size_t)wave * (NN / 4);
    const v4f* cs4 = ((const v4f*)colsum) + (size_t)b * (NN / 4);

    float q[16];
    float s = 0.0f;
    #pragma unroll
    for (int k = 0; k < 4; ++k) {
        const v4f p = p4[lane + 32 * k];
        const v4f c = cs4[lane + 32 * k];
        #pragma unroll
        for (int cc = 0; cc < 4; ++cc) {
            const float qv = p[cc] / c[cc];
            q[4 * k + cc] = qv;
            s += qv;
        }
    }
    s = waveSum(s);
    const float inv = 1.0f / s;
    #pragma unroll
    for (int k = 0; k < 4; ++k) {
        v4f o;
        #pragma unroll
        for (int cc = 0; cc < 4; ++cc) o[cc] = q[4 * k + cc] * inv;
        p4[lane + 32 * k] = o;
    }
}

__global__ void gs_init_softmax(const float* __restrict__ logits,
                                const float* __restrict__ u,
                                float* __restrict__ P) {
    const int wave = (blockIdx.x * blockDim.x + threadIdx.x) >> 5;
    const int lane = threadIdx.x & 31;
    const size_t row4 = (size_t)wave * (NN / 4);

    const v4f* lg4 = ((const v4f*)logits) + row4;
    const v4f* uu4 = ((const v4f*)u) + row4;
    v4f* p4 = ((v4f*)P) + row4;

    float y[16];
    float m = -3.402823466e38f;
    #pragma unroll
    for (int k = 0; k < 4; ++k) {
        const v4f L = __builtin_nontemporal_load(&lg4[lane + 32 * k]);
        const v4f U = __builtin_nontemporal_load(&uu4[lane + 32 * k]);
        #pragma unroll
        for (int c = 0; c < 4; ++c) {
            const float eps = -__logf(-__logf(U[c]));
            const float yv  = (L[c] + GAMMA * eps) * TEMP_INV;
            y[4 * k + c] = yv;
            m = fmaxf(m, yv);
        }
    }
    m = waveMax(m);
    float s = 0.0f;
    #pragma unroll
    for (int i = 0; i < 16; ++i) {
        const float e = __expf(y[i] - m);
        y[i] = e;
        s += e;
    }
    s = waveSum(s);
    const float inv = 1.0f / s;
    #pragma unroll
    for (int k = 0; k < 4; ++k) {
        v4f o;
        #pragma unroll
        for (int c = 0; c < 4; ++c) o[c] = y[4 * k + c] * inv;
        p4[lane + 32 * k] = o;
    }
}

// ---------------------------------------------------------------------------
// Launch. Preferred pipeline (needs ~8.1 MB scratch):
//   init(+partials of P0); 60 x { WMMA-finalize partials -> colsum ;
//                                 normalize(+partials of new P) }
// Per-iteration L2 traffic ~144 MB vs 192 MB for the naive 3-pass scheme.
// P (64 MB) stays resident in the 192 MB L2 for the whole loop.
// ---------------------------------------------------------------------------
extern "C" void kernel_launch(void* const* d_in, const int* in_sizes, int n_in,
                              void* d_out, int out_size, void* d_ws, size_t ws_size,
                              hipStream_t stream) {
    const float* logits = (const float*)d_in[0];
    const float* u      = (const float*)d_in[1];
    float* P            = (float*)d_out;

    float* colsum   = (float*)d_ws;                                   // 128 KB
    float* partials = (float*)((char*)d_ws + (size_t)NB * NN * sizeof(float));
    const size_t needPref = (size_t)NB * NN * sizeof(float)
                          + (size_t)NB * BLOCKS_PER_BATCH * NN * sizeof(float);

    const dim3 block(256);                           // 8 waves per block
    const dim3 gridRows((NB * NN) / ROWS_PER_BLOCK); // 4096 blocks, 1 wave/row
    const dim3 gridTiles((NB * 32 * 32) / 256);      // 256 blocks, 1 wave/16-col tile

    if (ws_size >= needPref) {
        gs_init_softmax_emit<<<gridRows, block, 0, stream>>>(logits, u, P, partials);
        for (int it = 0; it < ITERS; ++it) {
            gs_colsum_finalize_wmma<<<gridTiles, block, 0, stream>>>(partials, colsum);
            gs_normalize_emit<<<gridRows, block, 0, stream>>>(P, colsum, partials);
        }
    } else {
        gs_init_softmax<<<gridRows, block, 0, stream>>>(logits, u, P);
        for (int it = 0; it < ITERS; ++it) {
            gs_colsum_wmma<<<gridTiles, block, 0, stream>>>(P, colsum);
            gs_normalize<<<gridRows, block, 0, stream>>>(P, colsum);
        }
    }
}